// TemporalEMA_73194832658797
// MI455X (gfx1250) — compile-verified
//
#include <hip/hip_runtime.h>

// Temporal EMA with bilinear motion-vector warp.
// out = ALPHA * bilinear_warp(prev_hidden, mv) + (1-ALPHA) * current
// Shapes fixed by the reference: current/prev [4,16,1080,1920] f32, mv [4,2,1080,1920] f32.

#define ALPHA_F 0.85f

// Native clang vector type: required for __builtin_nontemporal_load/store,
// lowers to global_{load,store}_b128 with TH=NT.
typedef float v4f __attribute__((ext_vector_type(4)));

namespace {
constexpr int B   = 4;
constexpr int C   = 16;
constexpr int H   = 1080;
constexpr int W   = 1920;
constexpr int HW  = H * W;          // 2,073,600
constexpr int CHW = C * HW;         // 33,177,600
constexpr int PPT = 4;              // pixels (along w) per thread -> b128 vector ops
constexpr int W4  = W / PPT;        // 480, exact
constexpr int BLOCK = 160;          // 5 wave32s; W4 / BLOCK == 3 exactly
constexpr int GX  = W4 / BLOCK;     // 3
}

__global__ __launch_bounds__(BLOCK)
void temporal_ema_warp_kernel(const float* __restrict__ current,
                              const float* __restrict__ prev,
                              const float* __restrict__ mv,
                              float* __restrict__ out)
{
    // 3D grid: x = group-of-4 along w, y = row, z = batch.
    // No div/mod, no bounds check (exact factorization).
    const int g  = blockIdx.x * BLOCK + threadIdx.x;   // 0..479
    const int h  = blockIdx.y;                         // 0..1079 (scalar)
    const int b  = blockIdx.z;                         // 0..3    (scalar)
    const int w0 = g * PPT;

    const int rowBase = h * W + w0;                    // multiple of 4 -> 16B aligned

    // ---- Motion vectors: stream-once, vectorized, non-temporal ----
    const float* mvx_p = mv + (size_t)b * (2 * HW) + rowBase;
    const float* mvy_p = mvx_p + HW;
    const v4f mvx4 = __builtin_nontemporal_load((const v4f*)mvx_p);
    const v4f mvy4 = __builtin_nontemporal_load((const v4f*)mvy_p);

    const float mxa[PPT] = {mvx4.x, mvx4.y, mvx4.z, mvx4.w};
    const float mya[PPT] = {mvy4.x, mvy4.y, mvy4.z, mvy4.w};

    // Reference math collapses to:
    //   px = clamp(w - mvx*(W-1)/W, 0, W-1),  py = clamp(h - mvy*(H-1)/H, 0, H-1)
    const float sxw = (float)(W - 1) / (float)W;
    const float syh = (float)(H - 1) / (float)H;
    const float hf  = (float)h;

    int   i00[PPT], i01[PPT], i10[PPT], i11[PPT];
    float w00[PPT], w01[PPT], w10[PPT], w11[PPT];

#pragma unroll
    for (int j = 0; j < PPT; ++j) {
        float px = fminf(fmaxf((float)(w0 + j) - mxa[j] * sxw, 0.0f), (float)(W - 1));
        float py = fminf(fmaxf(hf              - mya[j] * syh, 0.0f), (float)(H - 1));
        float x0f = floorf(px);
        float y0f = floorf(py);
        float fx = px - x0f;
        float fy = py - y0f;
        int x0 = (int)x0f;
        int y0 = (int)y0f;
        int x1 = min(x0 + 1, W - 1);
        int y1 = min(y0 + 1, H - 1);
        int r0 = y0 * W;
        int r1 = y1 * W;
        i00[j] = r0 + x0;  i01[j] = r0 + x1;
        i10[j] = r1 + x0;  i11[j] = r1 + x1;
        float fxy = fx * fy;
        w11[j] = fxy;
        w10[j] = fy - fxy;
        w01[j] = fx - fxy;
        w00[j] = 1.0f - fx - fy + fxy;
    }

    const size_t base  = (size_t)b * CHW + rowBase;
    const float* curp  = current + base;
    const float* prvb  = prev + (size_t)b * CHW;
    float*       outp  = out + base;

    // Channel loop: weights/indices amortized over all 16 channels.
#pragma unroll 2
    for (int c = 0; c < C; ++c) {
        const float* pp = prvb + (size_t)c * HW;

        // Prefetch next channel's tap neighborhood (planes are 8.3 MB apart;
        // stride prefetchers won't catch this) -> global_prefetch_b8.
        if (c + 1 < C) {
            __builtin_prefetch(pp + HW + i00[0], 0, 3);
        }

        // Streamed current: b128, non-temporal (keep L2 for prev_hidden taps).
        const v4f cur = __builtin_nontemporal_load(
            (const v4f*)(curp + (size_t)c * HW));
        const float cura[PPT] = {cur.x, cur.y, cur.z, cur.w};

        float r[PPT];
#pragma unroll
        for (int j = 0; j < PPT; ++j) {
            const float v00 = pp[i00[j]];
            const float v01 = pp[i01[j]];
            const float v10 = pp[i10[j]];
            const float v11 = pp[i11[j]];
            float acc = w00[j] * v00;
            acc = fmaf(w01[j], v01, acc);
            acc = fmaf(w10[j], v10, acc);
            acc = fmaf(w11[j], v11, acc);
            r[j] = fmaf(ALPHA_F, acc, (1.0f - ALPHA_F) * cura[j]);
        }

        v4f o;
        o.x = r[0]; o.y = r[1]; o.z = r[2]; o.w = r[3];
        __builtin_nontemporal_store(o, (v4f*)(outp + (size_t)c * HW));
    }
}

extern "C" void kernel_launch(void* const* d_in, const int* in_sizes, int n_in,
                              void* d_out, int out_size, void* d_ws, size_t ws_size,
                              hipStream_t stream)
{
    (void)in_sizes; (void)n_in; (void)d_ws; (void)ws_size; (void)out_size;
    const float* current = (const float*)d_in[0];
    const float* prev    = (const float*)d_in[1];
    const float* mv      = (const float*)d_in[2];
    float*       out     = (float*)d_out;

    dim3 grid(GX, H, B);       // (3, 1080, 4)
    dim3 block(BLOCK, 1, 1);   // 160 threads = 5 wave32
    temporal_ema_warp_kernel<<<grid, block, 0, stream>>>(current, prev, mv, out);
}